// GNN_LP_24395414241389
// MI455X (gfx1250) — compile-verified
//
#include <hip/hip_runtime.h>

typedef float v2f __attribute__((ext_vector_type(2)));
typedef float v8f __attribute__((ext_vector_type(8)));

#define FEAT   128        // D == H == 128
#define EPSBN  1e-5f
#define KSLAB  64         // K rows of B staged per LDS pass
#define PAIRR  (KSLAB/2)  // 32 pair-rows
#define LPITCH 288        // floats per pair-row; 288 % 64 == 32 -> conflict-free b64 reads
#define CTILE  128        // columns per wave tile (8 x 16x16 WMMA tiles)

// ---------------------------------------------------------------------------
// 1) vectorized copy (agg = h, the GIN self-loop term)
// ---------------------------------------------------------------------------
__global__ void copy_f4_kernel(const float4* __restrict__ in, float4* __restrict__ out, int n4) {
    int i = blockIdx.x * blockDim.x + threadIdx.x;
    if (i < n4) out[i] = in[i];
}

// ---------------------------------------------------------------------------
// 2) scatter-add: agg[dst[e]] += h[src[e]]  (one wave32 per edge, float4/lane)
//    whole feature matrix (51 MB) is L2-resident, so this is L2-BW bound.
// ---------------------------------------------------------------------------
__global__ void scatter_add_kernel(const float* __restrict__ h,
                                   const int* __restrict__ src,
                                   const int* __restrict__ dst,
                                   float* __restrict__ agg, int E) {
    int gid  = blockIdx.x * blockDim.x + threadIdx.x;
    int e    = gid >> 5;
    int lane = gid & 31;
    if (e >= E) return;
    int s = src[e];
    int d = dst[e];
    const float4 v = *(const float4*)(h + (size_t)s * FEAT + lane * 4);
    float* p = agg + (size_t)d * FEAT + lane * 4;
    atomicAdd(p + 0, v.x);
    atomicAdd(p + 1, v.y);
    atomicAdd(p + 2, v.z);
    atomicAdd(p + 3, v.w);
}

// ---------------------------------------------------------------------------
// 3) fp32 WMMA GEMM, LDS-staged weights, 16x128 wave tile:
//      C[M,N] = act(A[M,K] @ B[K,N] + bias)
//    Block = 8 waves (256 thr): 8 row-tiles x 128 cols. B staged in LDS in
//    K-slabs of 64, (k,k+1)-paired so a B fragment is one ds_load_b64
//    (compiler fuses pairs into ds_load_2addr_b64). One A fragment feeds
//    8 WMMAs; A is software double-buffered across k-steps.
//    V_WMMA_F32_16X16X4_F32 layouts (wave32):
//      A 16x4 : lanes 0-15 rows M=0..15 {K=k,k+1}; lanes 16-31 {K=k+2,k+3}
//      B 4x16 : lanes 0-15 cols N=0..15 {K=k,k+1}; lanes 16-31 {K=k+2,k+3}
//      C/D    : v8f, lanes 0-15 M=r, N=lane; lanes 16-31 M=r+8
// ---------------------------------------------------------------------------
__device__ __forceinline__ void store_tile16(const v8f& acc, float* __restrict__ C,
                                             const float* __restrict__ bias,
                                             int row0, int col, int khalf, int l15,
                                             int N, int relu) {
    int c = col + l15;
    float bv = bias[c];
#pragma unroll
    for (int r = 0; r < 8; ++r) {
        float v = acc[r] + bv;
        if (relu) v = fmaxf(v, 0.0f);
        C[(size_t)(row0 + r + 8 * khalf) * N + c] = v;
    }
}

__global__ void __launch_bounds__(256)
gemm_bias_act_kernel(const float* __restrict__ A,
                     const float* __restrict__ B,
                     const float* __restrict__ bias,
                     float* __restrict__ C,
                     int M, int K, int N, int relu) {
    __shared__ float ldsB[PAIRR * LPITCH];          // 36 KB

    const int tid     = threadIdx.x;
    const int lane    = tid & 31;
    const int wave    = tid >> 5;
    const int rowTile = blockIdx.x * 8 + wave;
    const int row0    = rowTile * 16;
    const int col0    = blockIdx.y * CTILE;
    const int l15     = lane & 15;                  // A row / B col within tile
    const int khalf   = lane >> 4;                  // which half of the K=4 slab
    const bool active = (row0 < M);                 // wave-uniform tail guard

    v8f acc[8];
#pragma unroll
    for (int t = 0; t < 8; ++t) acc[t] = (v8f){};

    const float* Arow = A + (size_t)(row0 + l15) * K + 2 * khalf;

    for (int kbase = 0; kbase < K; kbase += KSLAB) {
        // ---- cooperative stage of B[kbase..kbase+64) x [col0..col0+128) ----
        for (int idx = tid; idx < KSLAB * CTILE; idx += 256) {
            int kk = idx >> 7;                      // 0..63 (coalesced along c)
            int c  = idx & (CTILE - 1);
            ldsB[(kk >> 1) * LPITCH + c * 2 + (kk & 1)] =
                B[(size_t)(kbase + kk) * N + col0 + c];
        }
        __syncthreads();

        if (active) {
            if (kbase + KSLAB < K)
                __builtin_prefetch(Arow + kbase + KSLAB, 0, 0);   // global_prefetch_b8
            v2f a = *(const v2f*)(Arow + kbase);                  // A double-buffer
#pragma unroll
            for (int k = 0; k < KSLAB; k += 4) {
                v2f a_nxt = a;
                if (k + 4 < KSLAB) a_nxt = *(const v2f*)(Arow + kbase + k + 4);
                const float* bp = &ldsB[((k >> 1) + khalf) * LPITCH + l15 * 2];
                v2f b0 = *(const v2f*)(bp);
                v2f b1 = *(const v2f*)(bp + 32);
                v2f b2 = *(const v2f*)(bp + 64);
                v2f b3 = *(const v2f*)(bp + 96);
                v2f b4 = *(const v2f*)(bp + 128);
                v2f b5 = *(const v2f*)(bp + 160);
                v2f b6 = *(const v2f*)(bp + 192);
                v2f b7 = *(const v2f*)(bp + 224);
                acc[0] = __builtin_amdgcn_wmma_f32_16x16x4_f32(false, a, false, b0, (short)0, acc[0], false, false);
                acc[1] = __builtin_amdgcn_wmma_f32_16x16x4_f32(false, a, false, b1, (short)0, acc[1], false, false);
                acc[2] = __builtin_amdgcn_wmma_f32_16x16x4_f32(false, a, false, b2, (short)0, acc[2], false, false);
                acc[3] = __builtin_amdgcn_wmma_f32_16x16x4_f32(false, a, false, b3, (short)0, acc[3], false, false);
                acc[4] = __builtin_amdgcn_wmma_f32_16x16x4_f32(false, a, false, b4, (short)0, acc[4], false, false);
                acc[5] = __builtin_amdgcn_wmma_f32_16x16x4_f32(false, a, false, b5, (short)0, acc[5], false, false);
                acc[6] = __builtin_amdgcn_wmma_f32_16x16x4_f32(false, a, false, b6, (short)0, acc[6], false, false);
                acc[7] = __builtin_amdgcn_wmma_f32_16x16x4_f32(false, a, false, b7, (short)0, acc[7], false, false);
                a = a_nxt;
            }
        }
        __syncthreads();
    }

    if (active) {
#pragma unroll
        for (int t = 0; t < 8; ++t)
            store_tile16(acc[t], C, bias, row0, col0 + 16 * t, khalf, l15, N, relu);
    }
}

// ---------------------------------------------------------------------------
// 4) BatchNorm (training-mode batch stats, biased variance)
// ---------------------------------------------------------------------------
__global__ void bn_zero_kernel(float* __restrict__ stats) {
    stats[threadIdx.x] = 0.0f;                      // launch with 256 threads
}

__global__ void bn_stats_kernel(const float* __restrict__ h,
                                float* __restrict__ stats, int N) {
    int c = threadIdx.x;                            // blockDim.x == FEAT
    float s = 0.0f, s2 = 0.0f;
    for (int row = blockIdx.x; row < N; row += gridDim.x) {
        float v = h[(size_t)row * FEAT + c];
        s  += v;
        s2 += v * v;
    }
    atomicAdd(&stats[c], s);
    atomicAdd(&stats[FEAT + c], s2);
}

__global__ void bn_apply_kernel(float* __restrict__ h,
                                const float* __restrict__ stats,
                                const float* __restrict__ gamma,
                                const float* __restrict__ beta,
                                int N, int relu) {
    int total = N * FEAT;
    float inv_n = 1.0f / (float)N;
    for (int idx = blockIdx.x * blockDim.x + threadIdx.x; idx < total;
         idx += gridDim.x * blockDim.x) {
        int c = idx & (FEAT - 1);
        float mean  = stats[c] * inv_n;
        float var   = stats[FEAT + c] * inv_n - mean * mean;
        float scale = gamma[c] * rsqrtf(var + EPSBN);
        float v = (h[idx] - mean) * scale + beta[c];
        if (relu) v = fmaxf(v, 0.0f);
        h[idx] = v;
    }
}

// ---------------------------------------------------------------------------
// 5) decode: out[i] = dot(z[e0[i]], z[e1[i]])  (one wave per label edge)
// ---------------------------------------------------------------------------
__global__ void decode_dot_kernel(const float* __restrict__ z,
                                  const int* __restrict__ e0,
                                  const int* __restrict__ e1,
                                  float* __restrict__ out, int EL) {
    int gid  = blockIdx.x * blockDim.x + threadIdx.x;
    int e    = gid >> 5;
    int lane = gid & 31;
    if (e >= EL) return;
    const float4 a = *(const float4*)(z + (size_t)e0[e] * FEAT + lane * 4);
    const float4 b = *(const float4*)(z + (size_t)e1[e] * FEAT + lane * 4);
    float p = a.x * b.x + a.y * b.y + a.z * b.z + a.w * b.w;
#pragma unroll
    for (int off = 16; off > 0; off >>= 1) p += __shfl_down(p, off, 32);
    if (lane == 0) out[e] = p;
}

// ---------------------------------------------------------------------------
// driver
// ---------------------------------------------------------------------------
extern "C" void kernel_launch(void* const* d_in, const int* in_sizes, int n_in,
                              void* d_out, int out_size, void* d_ws, size_t ws_size,
                              hipStream_t stream) {
    const float* x   = (const float*)d_in[0];
    const int*   ei  = (const int*)d_in[1];
    const int*   eli = (const int*)d_in[2];
    const float* W0a = (const float*)d_in[3];  const float* b0a = (const float*)d_in[4];
    const float* W0b = (const float*)d_in[5];  const float* b0b = (const float*)d_in[6];
    const float* W1a = (const float*)d_in[7];  const float* b1a = (const float*)d_in[8];
    const float* W1b = (const float*)d_in[9];  const float* b1b = (const float*)d_in[10];
    const float* g0  = (const float*)d_in[11]; const float* be0 = (const float*)d_in[12];
    const float* g1  = (const float*)d_in[13]; const float* be1 = (const float*)d_in[14];

    const int N  = in_sizes[0] / FEAT;   // 100000 nodes
    const int E  = in_sizes[1] / 2;      // 600000 edges
    const int EL = in_sizes[2] / 2;      // 200000 label edges
    const int* src = ei;        const int* dst = ei + E;
    const int* el0 = eli;       const int* el1 = eli + EL;

    float* B1    = (float*)d_ws;                  // [N,128]  agg buffer
    float* B2    = B1 + (size_t)N * FEAT;         // [N,256]  hidden buffer
    float* B3    = B2 + (size_t)N * 2 * FEAT;     // [N,128]  layer output
    float* stats = B3 + (size_t)N * FEAT;         // [256]    BN sum / sumsq

    const int n4        = N * FEAT / 4;
    const int copyBlks  = (n4 + 255) / 256;
    const int scatBlks  = ((E * 32) + 255) / 256;
    const int rowTiles  = (N + 15) / 16;
    const int gemmBlksX = (rowTiles + 7) / 8;     // 8 waves (256 thr) per block
    const int bnApplyBl = 4096;
    const int decBlks   = ((EL * 32) + 255) / 256;

    // ---- layer 0: GIN conv -> BN -> ReLU --------------------------------
    copy_f4_kernel<<<copyBlks, 256, 0, stream>>>((const float4*)x, (float4*)B1, n4);
    scatter_add_kernel<<<scatBlks, 256, 0, stream>>>(x, src, dst, B1, E);
    gemm_bias_act_kernel<<<dim3(gemmBlksX, 2), 256, 0, stream>>>(B1, W0a, b0a, B2,
                                                                 N, FEAT, 2 * FEAT, 1);
    gemm_bias_act_kernel<<<dim3(gemmBlksX, 1), 256, 0, stream>>>(B2, W0b, b0b, B3,
                                                                 N, 2 * FEAT, FEAT, 0);
    bn_zero_kernel<<<1, 256, 0, stream>>>(stats);
    bn_stats_kernel<<<1024, FEAT, 0, stream>>>(B3, stats, N);
    bn_apply_kernel<<<bnApplyBl, 256, 0, stream>>>(B3, stats, g0, be0, N, 1);

    // ---- layer 1: GIN conv -> BN (no activation) ------------------------
    copy_f4_kernel<<<copyBlks, 256, 0, stream>>>((const float4*)B3, (float4*)B1, n4);
    scatter_add_kernel<<<scatBlks, 256, 0, stream>>>(B3, src, dst, B1, E);
    gemm_bias_act_kernel<<<dim3(gemmBlksX, 2), 256, 0, stream>>>(B1, W1a, b1a, B2,
                                                                 N, FEAT, 2 * FEAT, 1);
    gemm_bias_act_kernel<<<dim3(gemmBlksX, 1), 256, 0, stream>>>(B2, W1b, b1b, B3,
                                                                 N, 2 * FEAT, FEAT, 0);
    bn_zero_kernel<<<1, 256, 0, stream>>>(stats);
    bn_stats_kernel<<<1024, FEAT, 0, stream>>>(B3, stats, N);
    bn_apply_kernel<<<bnApplyBl, 256, 0, stream>>>(B3, stats, g1, be1, N, 0);

    // ---- decode ----------------------------------------------------------
    decode_dot_kernel<<<decBlks, 256, 0, stream>>>(B3, el0, el1, (float*)d_out, EL);
}